// WaveletExtraction_42520176230823
// MI455X (gfx1250) — compile-verified
//
#include <hip/hip_runtime.h>

typedef float v2f __attribute__((ext_vector_type(2)));
typedef float v8f __attribute__((ext_vector_type(8)));

#define BN_EPS 1e-5f

// ---------------------------------------------------------------------------
// Haar DWT level: in (B,C,H,W) -> out (B,4C,H/2,W/2), channel order c*4+{ll,lh,hl,hh}
// ---------------------------------------------------------------------------
__global__ void haar_dwt2_kernel(const float* __restrict__ in, float* __restrict__ out,
                                 int B, int C, int H, int W) {
  const int H2 = H >> 1, W2 = W >> 1;
  const int total = B * C * H2 * W2;
  int idx = blockIdx.x * blockDim.x + threadIdx.x;
  if (idx >= total) return;
  int x = idx % W2; int t = idx / W2;
  int y = t % H2;   t /= H2;
  int c = t % C;    int b = t / C;
  const float* p = in + (((long)b * C + c) * H + 2 * y) * (long)W + 2 * x;
  float a  = p[0], bb = p[1], cc = p[W], dd = p[W + 1];
  float ll = (a + bb + cc + dd) * 0.5f;
  float lh = (a + bb - cc - dd) * 0.5f;
  float hl = (a - bb + cc - dd) * 0.5f;
  float hh = (a - bb - cc + dd) * 0.5f;
  float* q = out + (((long)b * (C * 4) + c * 4) * H2 + y) * (long)W2 + x;
  const long cs = (long)H2 * W2;
  q[0] = ll; q[cs] = lh; q[2 * cs] = hl; q[3 * cs] = hh;
}

// ---------------------------------------------------------------------------
// Pre-pack per-parity effective 2x2 weights as K-PAIRS so the B operand of
// V_WMMA_F32_16X16X4_F32 is one aligned b64 load per lane:
//   Wp2[((p*2*Ci + 2*ci + hp)*48 + co)*2 + e] = w[ci][co][3-(2a+py)][3-(2b+px)]
// with k = 4*ci + 2*hp + e, a = (k>>1)&1 = hp, b = k&1 = e, p = py*2+px.
// Also fold bias+BN: alpha = g*rsqrt(v+eps), betaE = (bias-m)*alpha + beta.
// ---------------------------------------------------------------------------
__global__ void pack_weights_kernel(const float* __restrict__ w, const float* __restrict__ bias,
                                    const float* __restrict__ g, const float* __restrict__ beta,
                                    const float* __restrict__ m, const float* __restrict__ v,
                                    float* __restrict__ Wp2, float* __restrict__ alpha,
                                    float* __restrict__ betaE, int Ci) {
  const int K = 4 * Ci;
  const int total = 4 * K * 48;
  int idx = blockIdx.x * blockDim.x + threadIdx.x;
  if (idx < 48) {
    float al = g[idx] * rsqrtf(v[idx] + BN_EPS);
    alpha[idx] = al;
    betaE[idx] = (bias[idx] - m[idx]) * al + beta[idx];
  }
  if (idx >= total) return;
  int co = idx % 48; int t = idx / 48;
  int k  = t % K;    int p = t / K;
  int ci = k >> 2;
  int hp = (k >> 1) & 1;     // which tap-pair (rows 0,1 vs 2,3 of this ci's K-block)
  int e  = k & 1;            // element within pair
  int py = p >> 1, px = p & 1;
  int ky = 2 * hp + py, kx = 2 * e + px;
  float val = w[(((long)(ci * 48 + co) * 4) + (3 - ky)) * 4 + (3 - kx)];
  Wp2[(((long)(p * 2 * Ci) + 2 * ci + hp) * 48 + co) * 2 + e] = val;
}

// ---------------------------------------------------------------------------
// Fused convT(4x4,s2,p1) + BN + ReLU via V_WMMA_F32_16X16X4_F32.
// Block = 4 waves, one wave per output parity (py,px); each wave does an
// M=32 x N=48 x K=4*Ci GEMM as 2 M-tiles x 3 N-tiles (6 accumulators), so the
// three b64 B-loads are amortized over 6 WMMAs per k-step.
// Input halo (Ci x 3 rows x 34 cols, zero-padded) is staged in LDS once, so
// the k-loop has NO predication: 4 ds_loads + 3 global b64 + 6 wmma.
// Epilogue: per-channel alpha/betaE + ReLU, restage 48x2x64 tile in the same
// LDS buffer, fully coalesced float4 stores.
// ---------------------------------------------------------------------------
__global__ __launch_bounds__(128)
void convT_bn_relu_wmma(const float* __restrict__ in, const float* __restrict__ Wp2,
                        const float* __restrict__ alpha, const float* __restrict__ betaE,
                        float* __restrict__ out, int Ci, int Hi, int Wi,
                        int CoutTotal, int coBase) {
  const int Wo   = 2 * Wi, Ho = 2 * Hi;
  const int tid  = threadIdx.x;
  const int wave = tid >> 5;
  const int lane = tid & 31;
  const int half = lane >> 4;     // selects K-rows {0,1} vs {2,3} of each ci block
  const int nn   = lane & 15;     // M index (A) / N index (B)
  const int py   = wave >> 1, px = wave & 1;
  const int b    = blockIdx.z;
  const int iy   = blockIdx.y;
  const int x0   = blockIdx.x * 32;           // 32 input columns per block

  // One LDS buffer reused: phase 1 = input halo (<= Ci*102 <= 4896 floats),
  // phase 2 = output tile (48co x 2rows x 64x = 6144 floats = 24 KB).
  __shared__ float4 smem4[1536];
  float* smem = (float*)smem4;

  // ---- Phase 1: stage zero-padded input halo: [ci][r=0..2][xx=0..33] ----
  {
    const int total = Ci * 102;               // 3*34 per channel
    for (int t = tid; t < total; t += 128) {
      int ci  = t / 102;
      int rem = t - ci * 102;
      int r   = rem / 34;
      int xx  = rem - r * 34;
      int y   = iy - 1 + r;
      int x   = x0 - 1 + xx;
      float val = 0.0f;
      if (y >= 0 && y < Hi && x >= 0 && x < Wi)
        val = in[(((long)b * Ci + ci) * Hi + y) * (long)Wi + x];
      smem[t] = val;
    }
  }
  __syncthreads();

  v8f cL0 = {}, cL1 = {}, cL2 = {};
  v8f cH0 = {}, cH1 = {}, cH2 = {};

  const v2f* wbase = (const v2f*)Wp2 + (long)((py * 2 + px) * 2 * Ci) * 48;

  for (int ci = 0; ci < Ci; ++ci) {
    // A operands from LDS (unpredicated; halo already zero-padded)
    const float* arow = smem + (ci * 3 + half + py) * 34 + px;
    v2f aL; aL.x = arow[nn];      aL.y = arow[nn + 1];
    v2f aH; aH.x = arow[nn + 16]; aH.y = arow[nn + 17];
    // B operands: one aligned b64 per N-tile, coalesced across lanes
    const v2f* wp = wbase + (long)(2 * ci + half) * 48;
    v2f b0 = wp[nn];
    v2f b1 = wp[nn + 16];
    v2f b2 = wp[nn + 32];
    cL0 = __builtin_amdgcn_wmma_f32_16x16x4_f32(false, aL, false, b0, (short)0, cL0, false, false);
    cH0 = __builtin_amdgcn_wmma_f32_16x16x4_f32(false, aH, false, b0, (short)0, cH0, false, false);
    cL1 = __builtin_amdgcn_wmma_f32_16x16x4_f32(false, aL, false, b1, (short)0, cL1, false, false);
    cH1 = __builtin_amdgcn_wmma_f32_16x16x4_f32(false, aH, false, b1, (short)0, cH1, false, false);
    cL2 = __builtin_amdgcn_wmma_f32_16x16x4_f32(false, aL, false, b2, (short)0, cL2, false, false);
    cH2 = __builtin_amdgcn_wmma_f32_16x16x4_f32(false, aH, false, b2, (short)0, cH2, false, false);
  }
  __syncthreads();   // done reading input tile; smem is reused for output

  // ---- Phase 2: BN+ReLU epilogue into LDS tile [co][row][64] ----
#pragma unroll
  for (int nt = 0; nt < 3; ++nt) {
    v8f accL = (nt == 0) ? cL0 : ((nt == 1) ? cL1 : cL2);
    v8f accH = (nt == 0) ? cH0 : ((nt == 1) ? cH1 : cH2);
    int   co = nt * 16 + nn;
    float al = alpha[co];
    float be = betaE[co];
    float* orow = smem + (co * 2 + py) * 64 + px;
#pragma unroll
    for (int r = 0; r < 8; ++r) {
      int m = r + 8 * half;   // C layout: VGPR r -> M=r (lanes<16), M=8+r (lanes>=16)
      orow[2 * m]        = fmaxf(fmaf(accL[r], al, be), 0.0f);
      orow[2 * (m + 16)] = fmaxf(fmaf(accH[r], al, be), 0.0f);
    }
  }
  __syncthreads();

  // ---- Phase 3: coalesced float4 stores (64 contiguous x per (co,row)) ----
  const long ob = ((long)b * CoutTotal + coBase) * Ho;
#pragma unroll
  for (int j = 0; j < 12; ++j) {
    int idx = tid + 128 * j;                  // 0..1535 float4s
    int x4  = idx & 15;
    int row = (idx >> 4) & 1;
    int co  = idx >> 5;
    float4 val = smem4[idx];
    float* dst = out + (ob + (long)co * Ho + (2 * iy + row)) * Wo + (2 * x0 + 4 * x4);
    *(float4*)dst = val;
  }
}

// ---------------------------------------------------------------------------
// Launch
// ---------------------------------------------------------------------------
extern "C" void kernel_launch(void* const* d_in, const int* in_sizes, int n_in,
                              void* d_out, int out_size, void* d_ws, size_t ws_size,
                              hipStream_t stream) {
  (void)in_sizes; (void)n_in; (void)out_size; (void)ws_size;
  const float* x      = (const float*)d_in[0];
  const float* w1     = (const float*)d_in[1];
  const float* b1     = (const float*)d_in[2];
  const float* g1     = (const float*)d_in[3];
  const float* beta1  = (const float*)d_in[4];
  const float* m1     = (const float*)d_in[5];
  const float* v1     = (const float*)d_in[6];
  const float* w2a    = (const float*)d_in[7];
  const float* b2a    = (const float*)d_in[8];
  const float* g2a    = (const float*)d_in[9];
  const float* beta2a = (const float*)d_in[10];
  const float* m2a    = (const float*)d_in[11];
  const float* v2a    = (const float*)d_in[12];
  const float* w2b    = (const float*)d_in[13];
  const float* b2b    = (const float*)d_in[14];
  const float* g2b    = (const float*)d_in[15];
  const float* beta2b = (const float*)d_in[16];
  const float* m2b    = (const float*)d_in[17];
  const float* v2b    = (const float*)d_in[18];
  float* out = (float*)d_out;

  const int B = 16, Cin = 3, H = 256, W = 256;

  // Workspace carve-up (all chunks stay 256B aligned)
  char*  ws    = (char*)d_ws;
  float* xl1   = (float*)ws;                 ws += (size_t)B * 12 * 128 * 128 * 4;  // 12.6 MB
  float* xl2   = (float*)ws;                 ws += (size_t)B * 48 * 64 * 64 * 4;    // 12.6 MB
  float* out2a = (float*)ws;                 ws += (size_t)B * 48 * 128 * 128 * 4;  // 50.3 MB
  float* Wp1   = (float*)ws;                 ws += (size_t)4 * 48 * 48 * 4;
  float* Wp2a  = (float*)ws;                 ws += (size_t)4 * 192 * 48 * 4;
  float* Wp2b  = (float*)ws;                 ws += (size_t)4 * 192 * 48 * 4;
  float* al1   = (float*)ws;                 ws += 64 * 4;
  float* be1   = (float*)ws;                 ws += 64 * 4;
  float* al2a  = (float*)ws;                 ws += 64 * 4;
  float* be2a  = (float*)ws;                 ws += 64 * 4;
  float* al2b  = (float*)ws;                 ws += 64 * 4;
  float* be2b  = (float*)ws;                 ws += 64 * 4;

  // Haar level 1: x -> xl1 (16,12,128,128)
  {
    int n = B * Cin * 128 * 128;
    haar_dwt2_kernel<<<(n + 255) / 256, 256, 0, stream>>>(x, xl1, B, Cin, H, W);
  }
  // Haar level 2: xl1 -> xl2 (16,48,64,64)
  {
    int n = B * 12 * 64 * 64;
    haar_dwt2_kernel<<<(n + 255) / 256, 256, 0, stream>>>(xl1, xl2, B, 12, 128, 128);
  }
  // Weight/BN packing
  {
    int n1 = 4 * (4 * 12) * 48;
    pack_weights_kernel<<<(n1 + 255) / 256, 256, 0, stream>>>(w1, b1, g1, beta1, m1, v1,
                                                              Wp1, al1, be1, 12);
    int n2 = 4 * (4 * 48) * 48;
    pack_weights_kernel<<<(n2 + 255) / 256, 256, 0, stream>>>(w2a, b2a, g2a, beta2a, m2a, v2a,
                                                              Wp2a, al2a, be2a, 48);
    pack_weights_kernel<<<(n2 + 255) / 256, 256, 0, stream>>>(w2b, b2b, g2b, beta2b, m2b, v2b,
                                                              Wp2b, al2b, be2b, 48);
  }
  // Branch 1: convT(xl1) -> out channels [0,48)
  {
    dim3 grid(128 / 32, 128, B), block(128);
    convT_bn_relu_wmma<<<grid, block, 0, stream>>>(xl1, Wp1, al1, be1, out,
                                                   12, 128, 128, 96, 0);
  }
  // Branch 2a: convT(xl2) -> out2a (16,48,128,128)
  {
    dim3 grid(64 / 32, 64, B), block(128);
    convT_bn_relu_wmma<<<grid, block, 0, stream>>>(xl2, Wp2a, al2a, be2a, out2a,
                                                   48, 64, 64, 48, 0);
  }
  // Branch 2b: convT(out2a) -> out channels [48,96)
  {
    dim3 grid(128 / 32, 128, B), block(128);
    convT_bn_relu_wmma<<<grid, block, 0, stream>>>(out2a, Wp2b, al2b, be2b, out,
                                                   48, 128, 128, 96, 48);
  }
}